// Involution_47510928228762
// MI455X (gfx1250) — compile-verified
//
#include <hip/hip_runtime.h>

typedef __attribute__((ext_vector_type(2))) float v2f;
typedef __attribute__((ext_vector_type(4))) float v4f;
typedef __attribute__((ext_vector_type(8))) float v8f;

#define CH    64
#define HH    128
#define WW    128
#define TMID  16     // reduced channels
#define NOUT  288    // K*K*GROUPS
#define MT    18     // 288/16 m-tiles for stage B
#define NWAVE 4      // waves per block

__global__ __launch_bounds__(128)
void involution_fused(const float* __restrict__ x,
                      const float* __restrict__ w1,
                      const float* __restrict__ b1,
                      const float* __restrict__ prelu_a,
                      const float* __restrict__ w2,
                      const float* __restrict__ b2,
                      float* __restrict__ out)
{
    // ---- LDS: W2 + b2 staged per block; T and wgt per wave ----
    __shared__ float s_w2[NOUT * TMID];      // 18 KB
    __shared__ float s_b2[NOUT];             // 1.2 KB
    __shared__ float s_T[NWAVE][16 * 16];    // 4 KB
    __shared__ float s_wgt[NWAVE][NOUT * 16];// 72 KB

    const int tid  = threadIdx.x;
    const int wv   = tid >> 5;
    const int lane = tid & 31;
    const int p    = lane & 15;   // pixel index within tile / matrix M-row
    const int hf   = lane >> 4;   // lane-half (selects K pair / row+8)
    const int h    = blockIdx.x;  // image row
    const int b    = blockIdx.y;  // batch

    // ---- async DMA of W2 (288x16) and b2 into LDS (ASYNCcnt path) ----
    {
        const unsigned long long w2base = (unsigned long long)(uintptr_t)w2;
        for (int i = tid; i < NOUT * TMID / 4; i += 128) {
            unsigned ldsa = (unsigned)(uintptr_t)&s_w2[4 * i];
            int      goff = 16 * i;
            asm volatile("global_load_async_to_lds_b128 %0, %1, %2"
                         :: "v"(ldsa), "v"(goff), "s"(w2base) : "memory");
        }
        const unsigned long long b2base = (unsigned long long)(uintptr_t)b2;
        for (int i = tid; i < NOUT / 4; i += 128) {
            unsigned ldsa = (unsigned)(uintptr_t)&s_b2[4 * i];
            int      goff = 16 * i;
            asm volatile("global_load_async_to_lds_b128 %0, %1, %2"
                         :: "v"(ldsa), "v"(goff), "s"(b2base) : "memory");
        }
    }

    // ---- prefetch the 3 x rows this block consumes (~96 KB, reused ~10x) ----
    for (int j = tid; j < CH * 3 * 4; j += 128) {   // 128B granules, 4 per 512B row
        const int gi  = j & 3;
        const int cr  = j >> 2;       // 0..191
        const int dh  = cr % 3;
        const int c   = cr / 3;
        const int hhh = h + dh - 1;
        if ((unsigned)hhh < (unsigned)HH)
            __builtin_prefetch(x + ((size_t)(b * CH + c) * HH + hhh) * WW + gi * 32, 0, 3);
    }

    // per-lane W1 A-fragments for all 16 K-steps (row-major (16,64): contiguous pair)
    v2f w1f[16];
#pragma unroll
    for (int k0 = 0; k0 < 16; ++k0)
        w1f[k0] = *(const v2f*)(w1 + p * CH + 4 * k0 + 2 * hf);

    // b1 prepared in C/D layout: VGPR r holds row r + 8*hf
    v8f b1p;
#pragma unroll
    for (int r = 0; r < 8; ++r) b1p[r] = b1[r + 8 * hf];

    const float alpha = prelu_a[0];

    asm volatile("s_wait_asynccnt 0" ::: "memory");  // our async LDS writes landed
    __syncthreads();                                  // ...and everyone else's

    // each wave walks 2 of the 8 tiles in this 128-wide row
    for (int t = wv; t < WW / 16; t += NWAVE) {
        const int w0  = t * 16;
        const int col = w0 + p;

        // ---- stage A: T(16x16) = W1(16x64) * X(64x16pix) + b1 ----
        // two independent WMMA accumulator chains for matrix-pipe ILP
        v8f ta = b1p;
        v8f tb = {};
#pragma unroll
        for (int k0 = 0; k0 < 16; k0 += 2) {
            const int c0 = 4 * k0 + 2 * hf;
            const float* xb0 = x + (((b * CH + c0) * HH + h) * WW + col);
            v2f bf0, bf1;
            bf0[0] = xb0[0];
            bf0[1] = xb0[HH * WW];
            bf1[0] = xb0[4 * HH * WW];
            bf1[1] = xb0[5 * HH * WW];
            ta = __builtin_amdgcn_wmma_f32_16x16x4_f32(
                false, w1f[k0], false, bf0, (short)0, ta, false, false);
            tb = __builtin_amdgcn_wmma_f32_16x16x4_f32(
                false, w1f[k0 + 1], false, bf1, (short)0, tb, false, false);
        }
        // merge + PReLU, then T -> LDS as [k][pixel]
#pragma unroll
        for (int r = 0; r < 8; ++r) {
            float v = ta[r] + tb[r];
            v = v >= 0.f ? v : alpha * v;
            s_T[wv][(r + 8 * hf) * 16 + p] = v;
        }

        // re-read T as stage-B B-fragments (K=16 -> 4 K-steps)
        v2f tB[4];
#pragma unroll
        for (int k0 = 0; k0 < 4; ++k0) {
            tB[k0][0] = s_T[wv][(4 * k0 + 2 * hf) * 16 + p];
            tB[k0][1] = s_T[wv][(4 * k0 + 2 * hf + 1) * 16 + p];
        }

        // ---- stage B: WGT(288x16pix) = W2(288x16) * T + b2, 18 m-tiles x 4 WMMA ----
#pragma unroll 2
        for (int m = 0; m < MT; ++m) {
            v4f q0 = *(const v4f*)&s_b2[16 * m + 8 * hf];      // ds_load_b128
            v4f q1 = *(const v4f*)&s_b2[16 * m + 8 * hf + 4];
            v8f c2;
#pragma unroll
            for (int r = 0; r < 4; ++r) { c2[r] = q0[r]; c2[r + 4] = q1[r]; }
#pragma unroll
            for (int k0 = 0; k0 < 4; ++k0) {
                v2f afrag = *(const v2f*)&s_w2[(16 * m + p) * TMID + 4 * k0 + 2 * hf];
                c2 = __builtin_amdgcn_wmma_f32_16x16x4_f32(
                    false, afrag, false, tB[k0], (short)0, c2, false, false);
            }
#pragma unroll
            for (int r = 0; r < 8; ++r)
                s_wgt[wv][(16 * m + r + 8 * hf) * 16 + p] = c2[r];
        }

        // ---- stage C: 3x3 data-dependent gather, packed over the channel pair ----
        // lane handles pixel `col`, groups [16*hf, 16*hf+16); channel pair shares wk
        for (int g2 = 0; g2 < 16; ++g2) {
            const int g = 16 * hf + g2;
            float wk[9];
#pragma unroll
            for (int kk = 0; kk < 9; ++kk)
                wk[kk] = s_wgt[wv][(g * 9 + kk) * 16 + p];

            const int c0 = 2 * g;
            const float* xc = x + (size_t)(b * CH + c0) * HH * WW;
            v2f acc = {0.f, 0.f};
#pragma unroll
            for (int dh = -1; dh <= 1; ++dh) {
                const int hhh = h + dh;
                const bool rok = (unsigned)hhh < (unsigned)HH;
                const float* xr = xc + hhh * WW;
#pragma unroll
                for (int dw = -1; dw <= 1; ++dw) {
                    const int cc = col + dw;
                    const bool ok = rok && ((unsigned)cc < (unsigned)WW);
                    const float wkv = wk[(dh + 1) * 3 + (dw + 1)];
                    v2f xv;
                    xv[0] = ok ? xr[cc] : 0.f;            // channel c0
                    xv[1] = ok ? xr[HH * WW + cc] : 0.f;  // channel c0+1
                    v2f wv2 = {wkv, wkv};
                    acc = __builtin_elementwise_fma(wv2, xv, acc);  // v_pk_fma_f32
                }
            }
            const int obase = ((b * CH + c0) * HH + h) * WW + col;
            out[obase]          = acc[0];
            out[obase + HH * WW] = acc[1];
        }
    }
}

extern "C" void kernel_launch(void* const* d_in, const int* in_sizes, int n_in,
                              void* d_out, int out_size, void* d_ws, size_t ws_size,
                              hipStream_t stream) {
    const float* x  = (const float*)d_in[0];
    const float* w1 = (const float*)d_in[1];
    const float* b1 = (const float*)d_in[2];
    const float* pa = (const float*)d_in[3];
    const float* w2 = (const float*)d_in[4];
    const float* b2 = (const float*)d_in[5];
    float* out = (float*)d_out;
    (void)in_sizes; (void)n_in; (void)out_size; (void)d_ws; (void)ws_size;

    dim3 grid(HH, 8);   // one block per (row, batch)
    dim3 block(128);    // 4 waves; each wave owns two 16-pixel tiles
    involution_fused<<<grid, block, 0, stream>>>(x, w1, b1, pa, w2, b2, out);
}